// KalmanFilter_33414845563529
// MI455X (gfx1250) — compile-verified
//
#include <hip/hip_runtime.h>

typedef float v2f __attribute__((ext_vector_type(2)));
typedef float v8f __attribute__((ext_vector_type(8)));

#define KF_B 1024
#define KF_T 64
#define KF_D 16
#define KF_N 32

// Same-wave LDS ordering (lanes are lockstep in one wave32; just drain DScnt).
__device__ __forceinline__ void lds_sync() {
  asm volatile("s_wait_dscnt 0" ::: "memory");
}

// D[MT*16 x NT*16] = opA(A) * opB(B) (+ D if ACC), K = KQ*4, fp32 WMMA.
// All matrices row-major in LDS. TA/TB transpose via LDS addressing.
// Fragment layouts per CDNA5 ISA 7.12.2:
//   A 16x4 : lane<16 -> {K=k0,k0+1}, lane>=16 -> {K=k0+2,k0+3}, M = lane&15
//   B 4x16 : same K split across lane halves, N = lane&15
//   C 16x16: vgpr r -> M = r + 8*(lane>=16), N = lane&15
template<bool TA, bool TB, bool ACC, int MT, int NT, int KQ>
__device__ __forceinline__ void wmm(float* __restrict__ d, int ldd,
                                    const float* __restrict__ a, int lda,
                                    const float* __restrict__ b, int ldb,
                                    int lane) {
  const int hl  = lane >> 4;   // lane-half
  const int l16 = lane & 15;
  #pragma unroll
  for (int tm = 0; tm < MT; ++tm) {
    #pragma unroll
    for (int tn = 0; tn < NT; ++tn) {
      v8f c{};
      if (ACC) {
        #pragma unroll
        for (int r = 0; r < 8; ++r)
          c[r] = d[(tm * 16 + r + 8 * hl) * ldd + tn * 16 + l16];
      }
      #pragma unroll
      for (int q = 0; q < KQ; ++q) {
        const int k0 = q * 4 + 2 * hl;
        const int m  = tm * 16 + l16;
        const int n  = tn * 16 + l16;
        v2f af, bf;
        af.x = TA ? a[(k0 + 0) * lda + m] : a[m * lda + k0 + 0];
        af.y = TA ? a[(k0 + 1) * lda + m] : a[m * lda + k0 + 1];
        bf.x = TB ? b[n * ldb + k0 + 0] : b[(k0 + 0) * ldb + n];
        bf.y = TB ? b[n * ldb + k0 + 1] : b[(k0 + 1) * ldb + n];
        c = __builtin_amdgcn_wmma_f32_16x16x4_f32(false, af, false, bf,
                                                  (short)0, c, false, false);
      }
      #pragma unroll
      for (int r = 0; r < 8; ++r)
        d[(tm * 16 + r + 8 * hl) * ldd + tn * 16 + l16] = c[r];
    }
  }
}

__global__ __launch_bounds__(32) void kf_wmma_kernel(
    const float* __restrict__ z, const unsigned char* __restrict__ maskp,
    const float* __restrict__ dtp, const float* __restrict__ Fg,
    const float* __restrict__ preQ, const float* __restrict__ Hg,
    const float* __restrict__ preR, float* __restrict__ out)
{
  __shared__ float Fsh[KF_N * KF_N];   // F
  __shared__ float Hsh[KF_D * KF_N];   // H
  __shared__ float q2sh[KF_N];         // pre_Q^2 (diag of Qm)
  __shared__ float prsh[KF_D];         // pre_R
  __shared__ float Pm[KF_N * KF_N];    // covariance (persistent)
  __shared__ float Xm[KF_N * KF_N];    // X = F*dt ; later IKH
  __shared__ float Am[KF_N * KF_N];    // A = expm(X)
  __shared__ float B0[KF_N * KF_N];    // ping-pong / scratch
  __shared__ float B1[KF_N * KF_N];    // ping-pong / scratch
  __shared__ float W0[KF_N * KF_N];    // (-X^T)^k / k! ping-pong ; later Lq
  __shared__ float W1[KF_N * KF_N];
  __shared__ float PH[KF_N * KF_N];    // Phi12 accumulator
  __shared__ float Hm[KF_D * KF_N];    // masked H
  __shared__ float HP[KF_D * KF_N];    // Hm @ P
  __shared__ float Kt[KF_D * KF_N];    // K^T = S^{-1} (Hm P)
  __shared__ float Sm[KF_D * KF_D];    // innovation cov / Cholesky factor
  __shared__ float xsh[KF_N];
  __shared__ float vsh[KF_D];
  __shared__ float mbf[KF_D];
  __shared__ float r2sh[KF_D];

  const int lane = threadIdx.x;
  const int b = blockIdx.x;

  // ---- one-time loads + state init ----
  for (int r = 0; r < KF_N; ++r) {
    Fsh[r * KF_N + lane] = Fg[r * KF_N + lane];
    Pm[r * KF_N + lane] = (r == lane) ? 1.f : 0.f;
  }
  for (int r = 0; r < KF_D; ++r) Hsh[r * KF_N + lane] = Hg[r * KF_N + lane];
  { const float q = preQ[lane]; q2sh[lane] = q * q; }
  if (lane < KF_D) prsh[lane] = preR[lane];
  xsh[lane] = 0.f;
  lds_sync();

  for (int t = 0; t < KF_T; ++t) {
    const float dtv = dtp[b * KF_T + t];

    // ======== discretize: A = expm(F*dt), 6-term Taylor (||F*dt|| ~ 0.1) ====
    for (int r = 0; r < KF_N; ++r) {
      const int idx = r * KF_N + lane;
      const float xv = Fsh[idx] * dtv;
      Xm[idx] = xv;
      Am[idx] = xv + ((r == lane) ? 1.f : 0.f);
      B0[idx] = xv;                       // running term X^k / k!
    }
    lds_sync();
    {
      float* cur = B0; float* nxt = B1;
      for (int k = 2; k <= 6; ++k) {
        wmm<false, false, false, 2, 2, 8>(nxt, KF_N, cur, KF_N, Xm, KF_N, lane);
        lds_sync();
        const float invk = 1.f / (float)k;
        for (int r = 0; r < KF_N; ++r) {
          const int idx = r * KF_N + lane;
          const float tv = nxt[idx] * invk;
          nxt[idx] = tv;
          Am[idx] += tv;
        }
        lds_sync();
        float* tp = cur; cur = nxt; nxt = tp;
      }
    }

    // ======== Lq via Van-Loan: Phi12 recursion, Lq = Phi12 @ A^T ===========
    // u_1 = Q*dt (diag); u_k = (X u_{k-1} + Q*dt * w_{k-1}) / k
    // w_1 = -X^T       ; w_k = w_{k-1} (-X^T) / k
    for (int r = 0; r < KF_N; ++r) {
      const int idx = r * KF_N + lane;
      const float u1 = (r == lane) ? q2sh[r] * dtv : 0.f;
      B0[idx] = u1;
      PH[idx] = u1;
      W0[idx] = -Xm[lane * KF_N + r];
    }
    lds_sync();
    {
      float* uc = B0; float* un = B1;
      float* wc = W0; float* wn = W1;
      for (int k = 2; k <= 5; ++k) {
        wmm<false, false, false, 2, 2, 8>(un, KF_N, Xm, KF_N, uc, KF_N, lane);
        lds_sync();
        const float invk = 1.f / (float)k;
        for (int r = 0; r < KF_N; ++r) {
          const int idx = r * KF_N + lane;
          const float tv = (un[idx] + q2sh[r] * dtv * wc[idx]) * invk;
          un[idx] = tv;
          PH[idx] += tv;
        }
        lds_sync();
        wmm<false, true, false, 2, 2, 8>(wn, KF_N, wc, KF_N, Xm, KF_N, lane);
        lds_sync();
        for (int r = 0; r < KF_N; ++r) wn[r * KF_N + lane] *= -invk;
        lds_sync();
        float* tp;
        tp = uc; uc = un; un = tp;
        tp = wc; wc = wn; wn = tp;
      }
    }
    wmm<false, true, false, 2, 2, 8>(W0, KF_N, PH, KF_N, Am, KF_N, lane); // Lq
    lds_sync();

    // ======== predict: x = A x ; P = A P A^T + Lq =========================
    wmm<false, false, false, 2, 2, 8>(B0, KF_N, Am, KF_N, Pm, KF_N, lane); // A P
    lds_sync();
    {
      float s = 0.f;
      for (int j = 0; j < KF_N; ++j) s += Am[lane * KF_N + j] * xsh[j];
      xsh[lane] = s;
    }
    lds_sync();
    for (int r = 0; r < KF_N; ++r) Pm[r * KF_N + lane] = W0[r * KF_N + lane];
    lds_sync();
    wmm<false, true, true, 2, 2, 8>(Pm, KF_N, B0, KF_N, Am, KF_N, lane); // +=(AP)A^T
    lds_sync();

    // ======== update ======================================================
    if (lane < KF_D) {
      const unsigned char mb = maskp[(b * KF_T + t) * KF_D + lane];
      mbf[lane] = mb ? 1.f : 0.f;
      const float pr = mb ? prsh[lane] : 1.f;
      r2sh[lane] = pr * pr;
    }
    lds_sync();
    for (int r = 0; r < KF_D; ++r) Hm[r * KF_N + lane] = mbf[r] * Hsh[r * KF_N + lane];
    lds_sync();
    wmm<false, false, false, 1, 2, 8>(HP, KF_N, Hm, KF_N, Pm, KF_N, lane); // Hm P
    lds_sync();
    wmm<false, true, false, 1, 1, 8>(Sm, KF_D, HP, KF_N, Hm, KF_N, lane);  // S
    lds_sync();
    if (lane < KF_D) Sm[lane * KF_D + lane] += r2sh[lane];
    lds_sync();

    // in-place wave-cooperative Cholesky (lower) of S
    for (int k = 0; k < KF_D; ++k) {
      const float dv = sqrtf(Sm[k * KF_D + k]);
      if (lane == k) Sm[k * KF_D + k] = dv;
      if (lane > k && lane < KF_D) Sm[lane * KF_D + k] /= dv;
      lds_sync();
      if (lane > k && lane < KF_D) {
        const float lik = Sm[lane * KF_D + k];
        for (int j = k + 1; j <= lane; ++j)
          Sm[lane * KF_D + j] -= lik * Sm[j * KF_D + k];
      }
      lds_sync();
    }

    // Kt = S^{-1} HP via fwd/back substitution, one LDS column per lane
    {
      float yv[KF_D];
      #pragma unroll
      for (int i = 0; i < KF_D; ++i) {
        float s = HP[i * KF_N + lane];
        #pragma unroll
        for (int j = 0; j < KF_D; ++j)
          if (j < i) s -= Sm[i * KF_D + j] * yv[j];
        yv[i] = s / Sm[i * KF_D + i];
      }
      float zv[KF_D];
      #pragma unroll
      for (int i = KF_D - 1; i >= 0; --i) {
        float s = yv[i];
        #pragma unroll
        for (int j = 0; j < KF_D; ++j)
          if (j > i) s -= Sm[j * KF_D + i] * zv[j];
        zv[i] = s / Sm[i * KF_D + i];
      }
      #pragma unroll
      for (int i = 0; i < KF_D; ++i) Kt[i * KF_N + lane] = zv[i];
    }
    lds_sync();

    // innovation + state update: v = z*m - Hm x ; x += K v
    if (lane < KF_D) {
      float s = z[(b * KF_T + t) * KF_D + lane] * mbf[lane];
      for (int j = 0; j < KF_N; ++j) s -= Hm[lane * KF_N + j] * xsh[j];
      vsh[lane] = s;
    }
    lds_sync();
    {
      float s = xsh[lane];
      for (int d = 0; d < KF_D; ++d) s += Kt[d * KF_N + lane] * vsh[d];
      xsh[lane] = s;
    }
    lds_sync();

    // Joseph form: IKH = I - K Hm ; P = IKH P IKH^T + K Rm K^T
    wmm<true, false, false, 2, 2, 4>(Xm, KF_N, Kt, KF_N, Hm, KF_N, lane);  // K Hm
    lds_sync();
    for (int r = 0; r < KF_N; ++r) {
      const int idx = r * KF_N + lane;
      Xm[idx] = ((r == lane) ? 1.f : 0.f) - Xm[idx];
    }
    lds_sync();
    wmm<false, false, false, 2, 2, 8>(B0, KF_N, Xm, KF_N, Pm, KF_N, lane); // IKH P
    lds_sync();
    for (int d = 0; d < KF_D; ++d)                       // KR = K diag(r2), 32x16
      B1[lane * KF_D + d] = Kt[d * KF_N + lane] * r2sh[d];
    lds_sync();
    wmm<false, false, false, 2, 2, 4>(Pm, KF_N, B1, KF_D, Kt, KF_N, lane); // K Rm K^T
    lds_sync();
    wmm<false, true, true, 2, 2, 8>(Pm, KF_N, B0, KF_N, Xm, KF_N, lane);   // + (IKH P) IKH^T
    lds_sync();

    // y = H x
    if (lane < KF_D) {
      float s = 0.f;
      for (int j = 0; j < KF_N; ++j) s += Hsh[lane * KF_N + j] * xsh[j];
      out[(b * KF_T + t) * KF_D + lane] = s;
    }
    lds_sync();
  }
}

extern "C" void kernel_launch(void* const* d_in, const int* in_sizes, int n_in,
                              void* d_out, int out_size, void* d_ws, size_t ws_size,
                              hipStream_t stream) {
  (void)in_sizes; (void)n_in; (void)d_ws; (void)ws_size; (void)out_size;
  const float* z           = (const float*)d_in[0];
  const unsigned char* mk  = (const unsigned char*)d_in[1];
  const float* dt          = (const float*)d_in[2];
  const float* F           = (const float*)d_in[3];
  const float* preQ        = (const float*)d_in[4];
  const float* H           = (const float*)d_in[5];
  const float* preR        = (const float*)d_in[6];
  float* out               = (float*)d_out;
  kf_wmma_kernel<<<dim3(KF_B), dim3(32), 0, stream>>>(z, mk, dt, F, preQ, H, preR, out);
}